// Top2Router_71571335020917
// MI455X (gfx1250) — compile-verified
//
#include <hip/hip_runtime.h>
#include <hip/hip_bf16.h>
#include <stdint.h>

// ---------------------------------------------------------------------------
// Top-2 MoE router: S=8192 tokens, E=64 experts, capacity=256.
// Output = [combine(8192,64,256) fp32 ; mask(8192,64,256) as 0/1 fp32], 1.07GB.
// Store-bandwidth bound (floor ~46us @ 23.3TB/s) => NT 128-bit stores for the
// bulk zero-fill; ranks via wave32 ballots + hierarchical scan; the scan
// stages histograms into LDS with the CDNA5 Tensor Data Mover.
// ---------------------------------------------------------------------------

#define S_TOK   8192
#define N_EXP   64
#define CAPTY   256
#define COMBINE_ELEMS ((size_t)S_TOK * N_EXP * CAPTY)   // 134,217,728

typedef unsigned int v4u __attribute__((ext_vector_type(4)));
typedef int          v4i __attribute__((ext_vector_type(4)));
typedef int          v8i __attribute__((ext_vector_type(8)));
typedef float        v4f __attribute__((ext_vector_type(4)));

// -------------------------------------------------------------------
// Kernel A: zero the 1.07GB output with non-temporal 128-bit stores.
// 4x unrolled so four b128 NT stores amortize each loop's exec-mask
// bookkeeping (each thread writes exactly 32 float4s at this config).
// -------------------------------------------------------------------
__global__ __launch_bounds__(256) void zerofill_kernel(v4f* __restrict__ p, size_t n4) {
    const size_t stride = (size_t)gridDim.x * blockDim.x;
    size_t i = (size_t)blockIdx.x * blockDim.x + threadIdx.x;
    v4f z = {0.f, 0.f, 0.f, 0.f};
    for (; i + 3 * stride < n4; i += 4 * stride) {
        __builtin_nontemporal_store(z, p + i);              // th:TH_STORE_NT
        __builtin_nontemporal_store(z, p + i + stride);
        __builtin_nontemporal_store(z, p + i + 2 * stride);
        __builtin_nontemporal_store(z, p + i + 3 * stride);
    }
    for (; i < n4; i += stride) {
        __builtin_nontemporal_store(z, p + i);
    }
}

// -------------------------------------------------------------------
// Kernel B: per-token softmax + top-2 + wave-local ranks/histograms.
// One wave (32 lanes) per 32 consecutive tokens; 64 logits per lane.
// Local rank within the wave via ballot+popc (wave32 native).
// -------------------------------------------------------------------
__global__ __launch_bounds__(32) void router_kernel(
    const float* __restrict__ in,
    int* __restrict__ idx1, int* __restrict__ idx2,
    float* __restrict__ w1o, float* __restrict__ w2o,
    int* __restrict__ lr1,  int* __restrict__ lr2,
    int* __restrict__ hist1, int* __restrict__ hist2)
{
    const int lane = threadIdx.x;                 // 0..31 (wave32)
    const int s    = blockIdx.x * 32 + lane;      // token id
    const float* row = in + (size_t)s * N_EXP;

    float v[N_EXP];
    const v4f* r4 = (const v4f*)row;
#pragma unroll
    for (int i = 0; i < N_EXP / 4; ++i) {
        v4f t = r4[i];
        v[4*i+0] = t.x; v[4*i+1] = t.y; v[4*i+2] = t.z; v[4*i+3] = t.w;
    }

    // softmax (fp32): exp(x - max) / sum
    float m = v[0];
#pragma unroll
    for (int i = 1; i < N_EXP; ++i) m = fmaxf(m, v[i]);
    float sum = 0.f;
#pragma unroll
    for (int i = 0; i < N_EXP; ++i) { v[i] = __expf(v[i] - m); sum += v[i]; }
    const float inv = 1.0f / sum;

    // top-1 (first occurrence of max — strict '>' keeps lowest index on ties)
    int   i1  = 0;
    float b1v = v[0];
#pragma unroll
    for (int i = 1; i < N_EXP; ++i) if (v[i] > b1v) { b1v = v[i]; i1 = i; }
    // top-2 with top-1 masked out
    int   i2  = (i1 == 0) ? 1 : 0;
    float b2v = v[i2];
#pragma unroll
    for (int i = 0; i < N_EXP; ++i) if (i != i1 && v[i] > b2v) { b2v = v[i]; i2 = i; }

    // wave-local ranks + per-block expert histograms via ballots
    unsigned m1 = 0u, m2 = 0u;
    int h1lo = 0, h1hi = 0, h2lo = 0, h2hi = 0;
#pragma unroll
    for (int e = 0; e < N_EXP; ++e) {
        unsigned bm1 = __builtin_amdgcn_ballot_w32(i1 == e);
        unsigned bm2 = __builtin_amdgcn_ballot_w32(i2 == e);
        if (i1 == e) m1 = bm1;
        if (i2 == e) m2 = bm2;
        if (e == lane)      { h1lo = __popc(bm1); h2lo = __popc(bm2); }
        if (e == lane + 32) { h1hi = __popc(bm1); h2hi = __popc(bm2); }
    }
    const unsigned lt = (1u << lane) - 1u;

    idx1[s] = i1;           idx2[s] = i2;
    w1o[s]  = b1v * inv;    w2o[s]  = b2v * inv;
    lr1[s]  = __popc(m1 & lt);
    lr2[s]  = __popc(m2 & lt);

    const int hb = blockIdx.x * N_EXP;            // 256 blocks x 64 experts
    hist1[hb + lane]      = h1lo;
    hist1[hb + 32 + lane] = h1hi;
    hist2[hb + lane]      = h2lo;
    hist2[hb + 32 + lane] = h2hi;
}

// -------------------------------------------------------------------
// TDM helper: DMA a 256x64 i32 tile (64KB) from global -> LDS offset 0.
// D# per CDNA5 ISA §8:
//   group0: count=1 | lds_addr=0 | global_addr | type=2
//   group1: data_size=4B, tensor_dim0=64, tensor_dim1=256,
//           tile_dim0=64, tile_dim1=256, tensor_dim0_stride=64
// -------------------------------------------------------------------
__device__ __forceinline__ void tdm_load_hist_to_lds(const void* gptr) {
    unsigned long long ga = (unsigned long long)(uintptr_t)gptr;
    v4u g0 = { 1u,                                        // count=1 (valid D#)
               0u,                                        // lds_addr = 0 (only LDS object)
               (unsigned)(ga & 0xFFFFFFFFull),
               (unsigned)((ga >> 32) & 0x01FFFFFFull) | 0x80000000u }; // addr[56:32]|type=2
    v8i g1 = { 0x00020000,    // workgroup_mask=0, data_size=2 (4B)
               0x00400000,    // tensor_dim0[15:0]=64 in bits[31:16]
               0x01000000,    // tensor_dim0[31:16]=0, tensor_dim1[15:0]=256
               0x00400000,    // tensor_dim1[31:16]=0, tile_dim0=64
               0x00000100,    // tile_dim1=256, tile_dim2=0
               64,            // tensor_dim0_stride[31:0]=64
               0, 0 };        // stride hi / dim1_stride (unused for 2D tile)
    v4i gz = { 0, 0, 0, 0 };
#if __has_include(<hip/amd_detail/amd_gfx1250_TDM.h>)
    v8i gz8 = { 0, 0, 0, 0, 0, 0, 0, 0 };
    __builtin_amdgcn_tensor_load_to_lds(g0, g1, gz, gz, gz8, 0);   // 6-arg toolchain
#else
    __builtin_amdgcn_tensor_load_to_lds(g0, g1, gz, gz, 0);        // 5-arg ROCm 7.2
#endif
}

// -------------------------------------------------------------------
// Kernel C: hierarchical scan. 64 threads (one per expert) scan the 256
// block histograms staged in LDS by the Tensor Data Mover.
//   offs1[b][e] = exclusive-cumsum over blocks of hist1
//   offs2[b][e] = total1[e] + exclusive-cumsum of hist2   (rank2 offset rule)
//
// DMA-visibility: (1) escape the LDS address as an INPUT-only asm operand
// with a "memory" clobber — AA must assume the asm (i.e. the TDM engine)
// stored to lh — while reads still index the original __shared__ array so
// they stay addrspace(3) and lower to ds_load_b32 (round-2's "+v" output
// pointer made them flat/vmem loads). (2) plant a never-taken store to lh
// behind an opaque SGPR zero so module-level "never stored" folding of the
// LDS global is impossible.
// -------------------------------------------------------------------
__global__ __launch_bounds__(64) void scan_kernel(
    const int* __restrict__ hist1, const int* __restrict__ hist2,
    int* __restrict__ offs1, int* __restrict__ offs2)
{
    __shared__ int lh[256 * N_EXP];               // 64 KB staging buffer
    const int e = threadIdx.x;                    // expert id 0..63

    // Opaque zero (compiler cannot prove it's 0) guards a dead store to lh.
    int opaque_zero;
    asm volatile("s_mov_b32 %0, 0" : "=s"(opaque_zero));
    if (opaque_zero) lh[e] = opaque_zero;         // never executes at runtime

    if (threadIdx.x < 32) {                       // wave 0 issues the DMA
        tdm_load_hist_to_lds(hist1);
        __builtin_amdgcn_s_wait_tensorcnt(0);     // s_wait_tensorcnt 0
    }
    __syncthreads();

    // "The DMA wrote lh": escape address, clobber memory, keep DS addressing.
    asm volatile("" : : "v"((int*)lh) : "memory");

    int run = 0;
#pragma unroll 8
    for (int b = 0; b < 256; ++b) {               // lh[b*64+e]: conflict-free banks
        offs1[b * N_EXP + e] = run;
        run += lh[b * N_EXP + e];                 // ds_load_b32
    }
    const int total1 = run;                       // pre-drop top-1 load per expert
    __syncthreads();                              // all reads of lh done

    if (threadIdx.x < 32) {
        tdm_load_hist_to_lds(hist2);              // reuse same LDS buffer
        __builtin_amdgcn_s_wait_tensorcnt(0);
    }
    __syncthreads();

    asm volatile("" : : "v"((int*)lh) : "memory");

    int run2 = total1;
#pragma unroll 8
    for (int b = 0; b < 256; ++b) {
        offs2[b * N_EXP + e] = run2;
        run2 += lh[b * N_EXP + e];                // ds_load_b32
    }
}

// -------------------------------------------------------------------
// Kernel D: scatter 2 weights + 2 mask bits per token into the output.
// top-1 expert != top-2 expert, so the two writes never collide.
// -------------------------------------------------------------------
__global__ __launch_bounds__(256) void scatter_kernel(
    const int* __restrict__ idx1, const int* __restrict__ idx2,
    const float* __restrict__ w1, const float* __restrict__ w2,
    const int* __restrict__ lr1,  const int* __restrict__ lr2,
    const int* __restrict__ offs1, const int* __restrict__ offs2,
    float* __restrict__ out)
{
    const int s = blockIdx.x * 256 + threadIdx.x;     // token id
    const int b = s >> 5;                             // wave-block id (32 tokens)
    const int e1 = idx1[s], e2 = idx2[s];
    const int r1 = offs1[b * N_EXP + e1] + lr1[s];
    const int r2 = offs2[b * N_EXP + e2] + lr2[s];

    float* __restrict__ combine = out;
    float* __restrict__ mask    = out + COMBINE_ELEMS;
    const size_t base = (size_t)s * (N_EXP * CAPTY);

    if (r1 < CAPTY) {
        size_t o = base + (size_t)e1 * CAPTY + r1;
        combine[o] = w1[s];
        mask[o]    = 1.0f;
    }
    if (r2 < CAPTY) {
        size_t o = base + (size_t)e2 * CAPTY + r2;
        combine[o] = w2[s];
        mask[o]    = 1.0f;
    }
}

// -------------------------------------------------------------------
extern "C" void kernel_launch(void* const* d_in, const int* in_sizes, int n_in,
                              void* d_out, int out_size, void* d_ws, size_t ws_size,
                              hipStream_t stream) {
    const float* in  = (const float*)d_in[0];    // [8192, 64] fp32
    float*       out = (float*)d_out;            // combine(134M) ++ mask(134M) fp32

    // Workspace layout (~448 KB total)
    int*   idx1  = (int*)d_ws;                   // [8192]
    int*   idx2  = idx1 + S_TOK;                 // [8192]
    float* w1    = (float*)(idx2 + S_TOK);       // [8192]
    float* w2    = w1 + S_TOK;                   // [8192]
    int*   lr1   = (int*)(w2 + S_TOK);           // [8192]
    int*   lr2   = lr1 + S_TOK;                  // [8192]
    int*   hist1 = lr2 + S_TOK;                  // [256*64]
    int*   hist2 = hist1 + 256 * N_EXP;          // [256*64]
    int*   offs1 = hist2 + 256 * N_EXP;          // [256*64]
    int*   offs2 = offs1 + 256 * N_EXP;          // [256*64]

    // A: zero 1.07 GB output (store-BW bound; NT b128 stores)
    const size_t n4 = (size_t)out_size / 4;      // float4 count
    zerofill_kernel<<<8192, 256, 0, stream>>>((v4f*)out, n4);

    // B: softmax + top-2 + wave-local ranks (256 waves, 32 tokens each)
    router_kernel<<<S_TOK / 32, 32, 0, stream>>>(in, idx1, idx2, w1, w2,
                                                 lr1, lr2, hist1, hist2);

    // C: cross-block exclusive scans per expert (TDM-staged LDS)
    scan_kernel<<<1, 64, 0, stream>>>(hist1, hist2, offs1, offs2);

    // D: scatter weights + mask
    scatter_kernel<<<S_TOK / 256, 256, 0, stream>>>(idx1, idx2, w1, w2,
                                                    lr1, lr2, offs1, offs2, out);
}